// ResGatedBlock_90761248899606
// MI455X (gfx1250) — compile-verified
//
#include <hip/hip_runtime.h>
#include <hip/hip_bf16.h>
#include <cstddef>

// ---------------------------------------------------------------------------
// Problem constants (match reference)
// ---------------------------------------------------------------------------
#define NNODES 20000
#define NEDGES 320000
#define DIN    256
#define H1DIM  256
#define H2DIM  128
#define DOUTD  64

typedef float v2f __attribute__((ext_vector_type(2)));
typedef float v8f __attribute__((ext_vector_type(8)));

// ---------------------------------------------------------------------------
// FP32 WMMA GEMM:  C[m][n] = sum_k A[m][k] * W[k][n] + bias[n]
// One wave (32 threads) per 16(M) x 64(N) strip; K stepped by 4 via
// V_WMMA_F32_16X16X4_F32.  A-fragment reused across the 4 N-tiles.
//
// ISA fragment layouts (CDNA5 ISA):
//   A 16x4 : lane(idx,half): v0 = A[idx][2*half], v1 = A[idx][2*half+1]
//   B 4x16 : lane(idx,half): v0 = B[2*half][idx], v1 = B[2*half+1][idx]
//   C 16x16: VGPR r: lanes0-15 -> C[r][idx], lanes16-31 -> C[r+8][idx]
// ---------------------------------------------------------------------------
__global__ __launch_bounds__(32)
void gemm_wmma_f32(const float* __restrict__ A,     // [Nrows x Din]
                   const float* __restrict__ W,     // [Din x Dout]
                   const float* __restrict__ bias,  // [Dout]
                   float* __restrict__ C,           // [Nrows x Dout]
                   int Din, int Dout)
{
    const int lane = threadIdx.x;        // 0..31 (wave32)
    const int half = lane >> 4;          // 0/1
    const int idx  = lane & 15;
    const int m0   = blockIdx.x * 16;
    const int n0   = blockIdx.y * 64;

    v8f acc0 = {}, acc1 = {}, acc2 = {}, acc3 = {};

    const float* arow = A + (size_t)(m0 + idx) * Din;

    for (int k0 = 0; k0 < Din; k0 += 4) {
        const int ka = k0 + 2 * half;
        v2f a;
        a.x = arow[ka];
        a.y = arow[ka + 1];

        const float* wr0 = W + (size_t)ka * Dout;        // row ka
        const float* wr1 = wr0 + Dout;                   // row ka+1

        v2f b0, b1, b2, b3;
        b0.x = wr0[n0 + 0  + idx];  b0.y = wr1[n0 + 0  + idx];
        b1.x = wr0[n0 + 16 + idx];  b1.y = wr1[n0 + 16 + idx];
        b2.x = wr0[n0 + 32 + idx];  b2.y = wr1[n0 + 32 + idx];
        b3.x = wr0[n0 + 48 + idx];  b3.y = wr1[n0 + 48 + idx];

        acc0 = __builtin_amdgcn_wmma_f32_16x16x4_f32(false, a, false, b0,
                                                     (short)0, acc0, false, false);
        acc1 = __builtin_amdgcn_wmma_f32_16x16x4_f32(false, a, false, b1,
                                                     (short)0, acc1, false, false);
        acc2 = __builtin_amdgcn_wmma_f32_16x16x4_f32(false, a, false, b2,
                                                     (short)0, acc2, false, false);
        acc3 = __builtin_amdgcn_wmma_f32_16x16x4_f32(false, a, false, b3,
                                                     (short)0, acc3, false, false);
    }

    // Write back: acc{t}[r] -> C[m0 + r + 8*half][n0 + t*16 + idx] + bias
    #pragma unroll
    for (int t = 0; t < 4; ++t) {
        const v8f acc = (t == 0) ? acc0 : (t == 1) ? acc1 : (t == 2) ? acc2 : acc3;
        const int nn = n0 + t * 16 + idx;
        const float bv = bias[nn];
        #pragma unroll
        for (int r = 0; r < 8; ++r) {
            C[(size_t)(m0 + r + 8 * half) * Dout + nn] = acc[r] + bv;
        }
    }
}

// ---------------------------------------------------------------------------
// Edge gate + scatter-add:
//   out[dst] += sigmoid(K[dst] + Q[src]) * V[src]   (per feature d)
// Thread per (edge, d). Feature dim contiguous across lanes -> coalesced.
// k/q/v arrays fit in the 192MB L2, so gathers and atomics are L2-resident.
// ---------------------------------------------------------------------------
__global__ __launch_bounds__(256)
void edge_gate_scatter(const int* __restrict__ src,
                       const int* __restrict__ dst,
                       const float* __restrict__ K,
                       const float* __restrict__ Q,
                       const float* __restrict__ V,
                       float* __restrict__ out,
                       int nEdges, int D, int logD)
{
    const long long tid = (long long)blockIdx.x * blockDim.x + threadIdx.x;
    if (tid >= (long long)nEdges * D) return;
    const int e = (int)(tid >> logD);
    const int d = (int)(tid & (D - 1));
    const int s = src[e];
    const int t = dst[e];
    const float g  = K[(size_t)t * D + d] + Q[(size_t)s * D + d];
    const float et = 1.0f / (1.0f + __expf(-g));
    atomicAdd(&out[(size_t)t * D + d], et * V[(size_t)s * D + d]);
}

// ---------------------------------------------------------------------------
// Row softmax over 64 columns: one wave per row, 2 elements per lane,
// butterfly reductions via __shfl_xor (wave32).
// ---------------------------------------------------------------------------
__global__ __launch_bounds__(256)
void softmax64(const float* __restrict__ in, float* __restrict__ out, int nRows)
{
    const int wavesPerBlock = blockDim.x >> 5;
    const int row  = blockIdx.x * wavesPerBlock + (threadIdx.x >> 5);
    const int lane = threadIdx.x & 31;
    if (row >= nRows) return;

    const float* p = in + (size_t)row * 64;
    float a = p[lane];
    float b = p[lane + 32];

    float m = fmaxf(a, b);
    #pragma unroll
    for (int off = 16; off > 0; off >>= 1)
        m = fmaxf(m, __shfl_xor(m, off, 32));

    const float ea = __expf(a - m);
    const float eb = __expf(b - m);
    float s = ea + eb;
    #pragma unroll
    for (int off = 16; off > 0; off >>= 1)
        s += __shfl_xor(s, off, 32);

    const float inv = 1.0f / s;
    float* o = out + (size_t)row * 64;
    o[lane]      = ea * inv;
    o[lane + 32] = eb * inv;
}

// ---------------------------------------------------------------------------
// Host-side orchestration
// ---------------------------------------------------------------------------
static inline void run_layer(const float* h, int Din_, int Dout_, int logD,
                             const float* kw, const float* kb,
                             const float* qw, const float* qb,
                             const float* vw, const float* vb,
                             const float* sw, const float* ob,
                             const int* src, const int* dst,
                             float* Kb, float* Qb, float* Vb, float* outp,
                             hipStream_t stream)
{
    dim3 blk(32);
    dim3 grd(NNODES / 16, Dout_ / 64);
    gemm_wmma_f32<<<grd, blk, 0, stream>>>(h, kw, kb, Kb, Din_, Dout_);
    gemm_wmma_f32<<<grd, blk, 0, stream>>>(h, qw, qb, Qb, Din_, Dout_);
    gemm_wmma_f32<<<grd, blk, 0, stream>>>(h, vw, vb, Vb, Din_, Dout_);
    gemm_wmma_f32<<<grd, blk, 0, stream>>>(h, sw, ob, outp, Din_, Dout_);  // out = skip + bias

    const long long total = (long long)NEDGES * Dout_;
    const int eblocks = (int)((total + 255) / 256);
    edge_gate_scatter<<<eblocks, 256, 0, stream>>>(src, dst, Kb, Qb, Vb, outp,
                                                   NEDGES, Dout_, logD);
}

extern "C" void kernel_launch(void* const* d_in, const int* in_sizes, int n_in,
                              void* d_out, int out_size, void* d_ws, size_t ws_size,
                              hipStream_t stream)
{
    (void)in_sizes; (void)n_in; (void)out_size; (void)ws_size;

    const float* x   = (const float*)d_in[0];
    const int*   ei  = (const int*)d_in[1];
    const int*   src = ei;            // edge_index[0]
    const int*   dst = ei + NEDGES;   // edge_index[1]

    const float* k1w = (const float*)d_in[2];  const float* k1b = (const float*)d_in[3];
    const float* q1w = (const float*)d_in[4];  const float* q1b = (const float*)d_in[5];
    const float* v1w = (const float*)d_in[6];  const float* v1b = (const float*)d_in[7];
    const float* k2w = (const float*)d_in[8];  const float* k2b = (const float*)d_in[9];
    const float* q2w = (const float*)d_in[10]; const float* q2b = (const float*)d_in[11];
    const float* v2w = (const float*)d_in[12]; const float* v2b = (const float*)d_in[13];
    const float* k3w = (const float*)d_in[14]; const float* k3b = (const float*)d_in[15];
    const float* q3w = (const float*)d_in[16]; const float* q3b = (const float*)d_in[17];
    const float* v3w = (const float*)d_in[18]; const float* v3b = (const float*)d_in[19];
    const float* s1w = (const float*)d_in[20]; const float* b1  = (const float*)d_in[21];
    const float* s2w = (const float*)d_in[22]; const float* b2  = (const float*)d_in[23];
    const float* s3w = (const float*)d_in[24]; const float* b3  = (const float*)d_in[25];

    // Workspace layout (floats)
    float* ws = (float*)d_ws;
    float* h1 = ws;                                  // N * 256
    float* h2 = h1 + (size_t)NNODES * H1DIM;         // N * 128
    float* h3 = h2 + (size_t)NNODES * H2DIM;         // N * 64
    float* Kb = h3 + (size_t)NNODES * DOUTD;         // N * 256 (shared across layers)
    float* Qb = Kb + (size_t)NNODES * H1DIM;         // N * 256
    float* Vb = Qb + (size_t)NNODES * H1DIM;         // N * 256

    // Layer 1: 256 -> 256
    run_layer(x,  DIN,   H1DIM, 8, k1w, k1b, q1w, q1b, v1w, v1b, s1w, b1,
              src, dst, Kb, Qb, Vb, h1, stream);
    // Layer 2: 256 -> 128
    run_layer(h1, H1DIM, H2DIM, 7, k2w, k2b, q2w, q2b, v2w, v2b, s2w, b2,
              src, dst, Kb, Qb, Vb, h2, stream);
    // Layer 3: 128 -> 64
    run_layer(h2, H2DIM, DOUTD, 6, k3w, k3b, q3w, q3b, v3w, v3b, s3w, b3,
              src, dst, Kb, Qb, Vb, h3, stream);

    // Softmax over 64 columns -> d_out
    softmax64<<<(NNODES + 7) / 8, 256, 0, stream>>>(h3, (float*)d_out, NNODES);
}